// RegionLossV2_83648783057303
// MI455X (gfx1250) — compile-verified
//
#include <hip/hip_runtime.h>
#include <math.h>

// Problem constants (match reference)
#define B_  32
#define T_  50
#define H_  76
#define W_  76
#define N_  5
#define C_  20
#define K_  25          // 5 + C_
#define HW_ (H_*W_)     // 5776
#define NCH_ (N_*K_)    // 125
#define OBJ_SCALE  5.0f
#define NOBJ_SCALE 1.0f
#define IOU_THRESH 0.6f

typedef float v2f __attribute__((ext_vector_type(2)));
typedef float v8f __attribute__((ext_vector_type(8)));

// ---------------------------------------------------------------------------
// Kernel 0: reset owner map to -1 and zero the output accumulator.
// (Must run every call: harness does not re-poison between replays.)
// ---------------------------------------------------------------------------
__global__ void rl_init_kernel(int* __restrict__ owner, float* __restrict__ out, int n) {
    int i = blockIdx.x * blockDim.x + threadIdx.x;
    if (i < n) owner[i] = -1;
    if (i == 0) out[0] = 0.0f;
}

// ---------------------------------------------------------------------------
// Kernel 1: per-target prep. One thread per (b,t).
//  - anchor assignment via IoU of (0,0,tw,th) vs per-cell decoded (pw,ph)
//  - encoded target tvol, coord scale, class id
//  - last-duplicate-wins owner scatter-max (matches .at[cell].max(flat_t))
// ---------------------------------------------------------------------------
__global__ void rl_target_kernel(const float*  __restrict__ output,
                                 const float*  __restrict__ target,
                                 const float*  __restrict__ priors,
                                 int*          __restrict__ owner,
                                 float4*       __restrict__ tvolw,
                                 float*        __restrict__ scalew,
                                 int*          __restrict__ tclsw) {
    int idx = blockIdx.x * blockDim.x + threadIdx.x;
    if (idx >= B_ * T_) return;
    int b = idx / T_;

    const float* tg = target + (size_t)idx * 5;
    float cls = tg[0], tx = tg[1], ty = tg[2], tw = tg[3], th = tg[4];
    int ci = (int)floorf(tx * (float)W_);
    int cj = (int)floorf(ty * (float)H_);

    float maxI = -1.0f; int maxn = 0;
    #pragma unroll
    for (int n = 0; n < N_; ++n) {
        size_t base = ((size_t)(b * NCH_ + n * K_ + 2) * H_ + cj) * W_ + ci;
        float o2 = output[base];
        float o3 = output[base + HW_];
        float pw = priors[2*n]     * expf(o2) * (1.0f / W_);
        float ph = priors[2*n + 1] * expf(o3) * (1.0f / H_);
        // IoU of zero-centered boxes: intersection = min(tw,pw)*min(th,ph)
        float iw = fminf(tw, pw), ih = fminf(th, ph);
        float inter = iw * ih;
        float iou = inter / (tw * th + pw * ph - inter);
        if (iou > maxI) { maxI = iou; maxn = n; }   // first-max == argmax
    }

    float pvx = (maxI != 0.0f) ? priors[2*maxn]     : 0.0f;
    float pvy = (maxI != 0.0f) ? priors[2*maxn + 1] : 0.0f;
    float tvx = tx * (float)W_ - (float)ci;
    float tvy = ty * (float)H_ - (float)cj;
    float tvw = logf(tw * (float)W_ / pvx);
    float tvh = logf(th * (float)H_ / pvy);
    if (isnan(tvw)) tvw = 0.0f;   // reference only replaces NaN (keeps inf)
    if (isnan(tvh)) tvh = 0.0f;

    tvolw[idx]  = make_float4(tvx, tvy, tvw, tvh);
    scalew[idx] = 2.0f - tw * th;
    tclsw[idx]  = (int)floorf(cls);

    int cell = ((b * H_ + cj) * W_ + ci) * N_ + maxn;
    atomicMax(&owner[cell], idx);   // later t wins (flat id scatter-max)
}

// ---------------------------------------------------------------------------
// Kernel 2: main per-cell pass + WMMA wave reduction.
// grid = (ceil(HW/256), N, B), block = 256 (8 wave32s).
// Per-channel loads are contiguous across hw -> fully coalesced.
// Targets for batch b staged once per block into LDS (broadcast reads).
// ---------------------------------------------------------------------------
__global__ void __launch_bounds__(256)
rl_cell_kernel(const float*  __restrict__ output,
               const float*  __restrict__ target,
               const float*  __restrict__ priors,
               const int*    __restrict__ owner,
               const float4* __restrict__ tvolw,
               const float*  __restrict__ scalew,
               const int*    __restrict__ tclsw,
               float*        __restrict__ out) {
    __shared__ float st[T_ * 5];
    __shared__ float wsums[8];

    const int tid = threadIdx.x;
    const int n   = blockIdx.y;
    const int b   = blockIdx.z;
    const int hw  = blockIdx.x * blockDim.x + tid;

    if (tid < T_ * 5) st[tid] = target[(size_t)b * T_ * 5 + tid];
    __syncthreads();

    const float prw = priors[2*n], prh = priors[2*n + 1];

    float partial = 0.0f;
    if (hw < HW_) {
        const int h = hw / W_, w = hw % W_;
        const float* op = output + (size_t)(b * NCH_ + n * K_) * HW_ + hw;
        float o[K_];
        #pragma unroll
        for (int k = 0; k < K_; ++k) o[k] = op[(size_t)k * HW_];

        // decoded predicted box
        float px = ((float)w + o[0]) * (1.0f / W_);
        float py = ((float)h + o[1]) * (1.0f / H_);
        float pw = prw * expf(o[2]) * (1.0f / W_);
        float ph = prh * expf(o[3]) * (1.0f / H_);
        float pl = px - pw * 0.5f, pr = px + pw * 0.5f;
        float pt = py - ph * 0.5f, pb = py + ph * 0.5f;
        float pa = pw * ph;

        // max IoU vs all 50 targets (LDS broadcast)
        float maxiou = 0.0f;
        for (int t = 0; t < T_; ++t) {
            float bx = st[t*5+1], by = st[t*5+2], bw = st[t*5+3], bh = st[t*5+4];
            float iw = fmaxf(fminf(pr, bx + bw*0.5f) - fmaxf(pl, bx - bw*0.5f), 0.0f);
            float ih = fmaxf(fminf(pb, by + bh*0.5f) - fmaxf(pt, by - bh*0.5f), 0.0f);
            float inter = iw * ih;
            float iou = inter / (pa + bw*bh - inter);
            maxiou = fmaxf(maxiou, iou);
        }
        const bool ioumask = maxiou > IOU_THRESH;

        const int cell = (b * HW_ + hw) * N_ + n;
        const int o_   = owner[cell];
        const bool pmask = (o_ >= 0);
        const int ow = pmask ? o_ : 0;
        const float4 tv = tvolw[ow];
        const float  sc = scalew[ow];
        const int    cl = tclsw[ow];

        float ss = 0.0f;
        if (pmask) {
            float d0 = sc * (tv.x - o[0]);
            float d1 = sc * (tv.y - o[1]);
            float d2 = sc * (tv.z - o[2]);
            float d3 = sc * (tv.w - o[3]);
            ss += d0*d0 + d1*d1 + d2*d2 + d3*d3;
            float dobj = OBJ_SCALE * (1.0f - o[4]);
            ss += dobj * dobj;
            #pragma unroll
            for (int c = 0; c < C_; ++c) {
                float d = ((c == cl) ? 1.0f : 0.0f) - o[5 + c];
                ss += d * d;
            }
        } else {
            float dobj = ioumask ? 0.0f : (NOBJ_SCALE * (0.0f - o[4]));
            ss += dobj * dobj;
        }
        partial = ss;
    }

    // ---- wave32 reduction via V_WMMA_F32_16X16X4_F32 ----
    // A layout (16x4 f32): lanes 0-15 hold rows M, VGPR0=K0/VGPR1=K1;
    // lanes 16-31 hold K2/K3. With a={partial,0} and B=ones:
    //   D[m][*] = p[m] + p[m+16]  (row sums, replicated over columns)
    // Each lane sums its 8 D regs -> lanes 0-15 hold sum(m=0..7),
    // lanes 16-31 hold sum(m=8..15); one shfl_xor(16) gives the wave total.
    v2f a;  a[0] = partial; a[1] = 0.0f;
    v2f bo; bo[0] = 1.0f;   bo[1] = 1.0f;
    v8f cz = {0.0f, 0.0f, 0.0f, 0.0f, 0.0f, 0.0f, 0.0f, 0.0f};
    v8f d = __builtin_amdgcn_wmma_f32_16x16x4_f32(
        /*neg_a=*/false, a, /*neg_b=*/false, bo,
        /*c_mod=*/(short)0, cz, /*reuse_a=*/false, /*reuse_b=*/false);
    float s8 = d[0] + d[1] + d[2] + d[3] + d[4] + d[5] + d[6] + d[7];
    float wavesum = s8 + __shfl_xor(s8, 16, 32);

    const int lane = tid & 31, wv = tid >> 5;
    if (lane == 0) wsums[wv] = wavesum;
    __syncthreads();
    if (tid == 0) {
        float bs = 0.0f;
        #pragma unroll
        for (int i = 0; i < 8; ++i) bs += wsums[i];
        atomicAdd(out, bs);
    }
}

// ---------------------------------------------------------------------------
// Kernel 3: cost = sqrt(sum)^2  (mirror the reference's rounding)
// ---------------------------------------------------------------------------
__global__ void rl_finalize_kernel(float* out) {
    float r = sqrtf(out[0]);
    out[0] = r * r;
}

// ---------------------------------------------------------------------------
extern "C" void kernel_launch(void* const* d_in, const int* in_sizes, int n_in,
                              void* d_out, int out_size, void* d_ws, size_t ws_size,
                              hipStream_t stream) {
    const float* output = (const float*)d_in[0];   // [B,125,H,W] f32
    const float* target = (const float*)d_in[1];   // [B,T,5]     f32
    const float* priors = (const float*)d_in[2];   // [10]        f32
    float* out = (float*)d_out;

    char* ws = (char*)d_ws;
    const int ncell = B_ * HW_ * N_;               // 924,160
    int*    owner  = (int*)ws;
    size_t  off    = (size_t)ncell * sizeof(int);  // 3,696,640 B (16-aligned)
    float4* tvolw  = (float4*)(ws + off); off += (size_t)B_ * T_ * sizeof(float4);
    float*  scalew = (float*)(ws + off);  off += (size_t)B_ * T_ * sizeof(float);
    int*    tclsw  = (int*)(ws + off);    // total ~3.6 MB << ws_size

    rl_init_kernel<<<(ncell + 255) / 256, 256, 0, stream>>>(owner, out, ncell);
    rl_target_kernel<<<(B_ * T_ + 255) / 256, 256, 0, stream>>>(
        output, target, priors, owner, tvolw, scalew, tclsw);
    rl_cell_kernel<<<dim3((HW_ + 255) / 256, N_, B_), 256, 0, stream>>>(
        output, target, priors, owner, tvolw, scalew, tclsw, out);
    rl_finalize_kernel<<<1, 1, 0, stream>>>(out);
}